// EPDetector_39616778338616
// MI455X (gfx1250) — compile-verified
//
#include <hip/hip_runtime.h>

typedef float v2f __attribute__((ext_vector_type(2)));
typedef float v8f __attribute__((ext_vector_type(8)));

#define L_ITERS 10
#define BETA    0.9f
#define PREC    1e-6f

// Uniform-lane broadcast: lowers to v_readlane_b32 (SGPR result, no LDS, no waits)
__device__ __forceinline__ float bcast(float v, int srcLane) {
    return __int_as_float(__builtin_amdgcn_readlane(__float_as_int(v), srcLane));
}

// Fast reciprocal: v_rcp_f32 + one Newton-Raphson step (SPD system, well conditioned)
__device__ __forceinline__ float frcp(float x) {
    float r = __builtin_amdgcn_rcpf(x);
    r = fmaf(r, fmaf(-x, r, 1.0f), r);
    return r;
}

__launch_bounds__(128, 1)
__global__ void ep_detect_kernel(const float* __restrict__ yr,
                                 const float* __restrict__ yi,
                                 const float* __restrict__ hr,
                                 const float* __restrict__ hi,
                                 const float* __restrict__ sdiag,
                                 float* __restrict__ out)
{
    __shared__ float sh_h[4][64 * 33];   // whitened real channel, 64x32 (+pad, odd stride -> conflict-free cols)
    __shared__ float sh_g[4][32 * 36];   // hth, 32x32 (+pad to 36 -> 16B-aligned rows for b128 loads)
    __shared__ float sh_y[4][64];        // whitened stacked y

    const int wid  = threadIdx.x >> 5;
    const int lane = threadIdx.x & 31;
    const int b    = blockIdx.x * 4 + wid;

    float* hb = sh_h[wid];
    float* gb = sh_g[wid];
    float* yb = sh_y[wid];

    // ---------------- load + whiten (lane = receive antenna m) ----------------
    const int m = lane;
    float w = rsqrtf(sdiag[b * 32 + m]);
    yb[m]      = yr[b * 32 + m] * w;
    yb[m + 32] = yi[b * 32 + m] * w;

    const float4* hr4 = (const float4*)(hr + (size_t)(b * 32 + m) * 16);
    const float4* hi4 = (const float4*)(hi + (size_t)(b * 32 + m) * 16);
#pragma unroll
    for (int q = 0; q < 4; ++q) {
        float4 r  = hr4[q];
        float4 ii = hi4[q];
        float rv[4] = { r.x, r.y, r.z, r.w };
        float iv[4] = { ii.x, ii.y, ii.z, ii.w };
#pragma unroll
        for (int t = 0; t < 4; ++t) {
            int c = q * 4 + t;
            float hrw = rv[t] * w, hiw = iv[t] * w;
            hb[m * 33 + c]             =  hrw;   // [ hr  -hi ]
            hb[m * 33 + 16 + c]        = -hiw;   // [ hi   hr ]
            hb[(m + 32) * 33 + c]      =  hiw;
            hb[(m + 32) * 33 + 16 + c] =  hrw;
        }
    }
    __syncthreads();

    // ---------------- hth = h^T h via V_WMMA_F32_16X16X4_F32 ----------------
    // A(i,k) = h[m0+k][16I+i], B(k,j) = h[m0+k][16J+j]  -> D = hth tile (I,J)
    // frag layout: lanes 0-15 i=lane (K=0/1 in .x/.y), lanes 16-31 (K=2/3)
    const int i16 = lane & 15;
    const int kh  = lane >> 4;
    v8f acc00 = {}; v8f acc01 = {}; v8f acc10 = {}; v8f acc11 = {};
#pragma unroll
    for (int kc = 0; kc < 16; ++kc) {
        int r0 = kc * 4 + kh * 2;
        v2f f0, f1;                       // column-block 0 and 1 fragments
        f0.x = hb[r0 * 33 + i16];
        f0.y = hb[(r0 + 1) * 33 + i16];
        f1.x = hb[r0 * 33 + 16 + i16];
        f1.y = hb[(r0 + 1) * 33 + 16 + i16];
        acc00 = __builtin_amdgcn_wmma_f32_16x16x4_f32(false, f0, false, f0, (short)0, acc00, false, false);
        acc01 = __builtin_amdgcn_wmma_f32_16x16x4_f32(false, f0, false, f1, (short)0, acc01, false, false);
        acc10 = __builtin_amdgcn_wmma_f32_16x16x4_f32(false, f1, false, f0, (short)0, acc10, false, false);
        acc11 = __builtin_amdgcn_wmma_f32_16x16x4_f32(false, f1, false, f1, (short)0, acc11, false, false);
    }
    // C/D layout: VGPR r, lanes 0-15 row r, lanes 16-31 row 8+r ; col = lane&15
#pragma unroll
    for (int r = 0; r < 8; ++r) {
        int rr = kh * 8 + r;
        gb[rr * 36 + i16]             = acc00[r];
        gb[rr * 36 + 16 + i16]        = acc01[r];
        gb[(16 + rr) * 36 + i16]      = acc10[r];
        gb[(16 + rr) * 36 + 16 + i16] = acc11[r];
    }
    __syncthreads();

    // ---------------- hty (lane j owns component j) + hth row to regs -------
    float bty = 0.f;
#pragma unroll 8
    for (int mm = 0; mm < 64; ++mm)
        bty = fmaf(hb[mm * 33 + lane], yb[mm], bty);

    float G[32];
#pragma unroll
    for (int j4 = 0; j4 < 8; ++j4) {     // 16B-aligned rows -> ds_load_b128
        float4 g4 = *(const float4*)(gb + lane * 36 + j4 * 4);
        G[j4 * 4 + 0] = g4.x; G[j4 * 4 + 1] = g4.y;
        G[j4 * 4 + 2] = g4.z; G[j4 * 4 + 3] = g4.w;
    }

    // ---------------- EP iterations, all in registers ----------------
    const float Pt[4] = { 0.31622776601683794f,  0.9486832980505138f,
                         -0.31622776601683794f, -0.9486832980505138f };
    float lam = 2.0f;   // 1/ES, ES = 0.5
    float gam = 0.0f;
    float lg0 = 0.f, lg1 = 0.f, lg2 = 0.f, lg3 = 0.f;

#pragma unroll 1
    for (int it = 0; it < L_ITERS; ++it) {
        // A = hth + 0.5*diag(lam)  (lane i owns row i)
        float a[32];
#pragma unroll
        for (int j = 0; j < 32; ++j)
            a[j] = (lane == j) ? (G[j] + 0.5f * lam) : G[j];

        // In-place Gauss-Jordan inversion (SPD, no pivoting).
        // k is compile-time constant after unroll -> v_readlane broadcasts.
#pragma unroll
        for (int k = 0; k < 32; ++k) {
            float akk  = bcast(a[k], k);
            float p    = frcp(akk);
            float fp   = a[k] * p;
            bool  isk  = (lane == k);
            float coef = isk ? p : -fp;
#pragma unroll
            for (int j = 0; j < 32; ++j) {
                if (j == k) continue;
                float rkj  = bcast(a[j], k);      // old pivot-row value (SGPR)
                float base = isk ? 0.f : a[j];
                a[j] = fmaf(coef, rkj, base);
            }
            a[k] = coef;
        }

        // mu_i = sum_j Ainv[i][j] * (hty_j + 0.5*gam_j) ; diag extraction
        float rhs = fmaf(0.5f, gam, bty);
        float mu = 0.f, diag = 0.f;
#pragma unroll
        for (int j = 0; j < 32; ++j) {
            float rj = bcast(rhs, j);
            mu   = fmaf(a[j], rj, mu);
            diag = (lane == j) ? a[j] : diag;
        }

        float sigma   = 0.5f * diag;
        float invSig  = frcp(sigma);
        float vobs    = frcp(invSig - lam);
        vobs = fmaxf(vobs, PREC);
        float xobs = vobs * fmaf(mu, invSig, -gam);

        float ivobs = frcp(vobs);
        float inv2v = 0.5f * ivobs;
        float d0 = xobs - Pt[0], d1 = xobs - Pt[1], d2 = xobs - Pt[2], d3 = xobs - Pt[3];
        lg0 = -d0 * d0 * inv2v; lg1 = -d1 * d1 * inv2v;
        lg2 = -d2 * d2 * inv2v; lg3 = -d3 * d3 * inv2v;

        float mx = fmaxf(fmaxf(lg0, lg1), fmaxf(lg2, lg3));
        float e0 = __expf(lg0 - mx), e1 = __expf(lg1 - mx);
        float e2 = __expf(lg2 - mx), e3 = __expf(lg3 - mx);
        float invZ = frcp(e0 + e1 + e2 + e3);
        float x = (Pt[0] * e0 + Pt[1] * e1 + Pt[2] * e2 + Pt[3] * e3) * invZ;
        float q0 = Pt[0] - x, q1 = Pt[1] - x, q2 = Pt[2] - x, q3 = Pt[3] - x;
        float v = (q0 * q0 * e0 + q1 * q1 * e1 + q2 * q2 * e2 + q3 * q3 * e3) * invZ;
        v = fmaxf(v, PREC);

        float iv   = frcp(v);
        float lamn = iv - ivobs;
        float gamn = x * iv - xobs * ivobs;
        bool  neg  = lamn < 0.f;
        float lamNew = neg ? lam : lamn;
        float gamNew = neg ? gam : gamn;
        lam = fmaf(1.0f - BETA, lamNew, BETA * lam);
        gam = fmaf(1.0f - BETA, gamNew, BETA * gam);
    }

    // ---------------- max-log LLR ----------------
    // Gray bits: pt0=[0,0] pt1=[0,1] pt2=[1,0] pt3=[1,1]
    float llrb0 = fmaxf(lg2, lg3) - fmaxf(lg0, lg1);   // MSB
    float llrb1 = fmaxf(lg1, lg3) - fmaxf(lg0, lg2);   // LSB
    int    k16 = lane & 15;            // complex symbol index
    int    p   = lane >> 4;            // 0 = real (pam1), 1 = imag (pam2)
    size_t base = ((size_t)b * 16 + k16) * 4;
    out[base + p]     = llrb0;
    out[base + 2 + p] = llrb1;
}

extern "C" void kernel_launch(void* const* d_in, const int* in_sizes, int n_in,
                              void* d_out, int out_size, void* d_ws, size_t ws_size,
                              hipStream_t stream) {
    const float* yr = (const float*)d_in[0];
    const float* yi = (const float*)d_in[1];
    const float* hr = (const float*)d_in[2];
    const float* hi = (const float*)d_in[3];
    const float* sd = (const float*)d_in[4];
    float* out = (float*)d_out;

    int B = in_sizes[0] / 32;          // yr is (B, M=32)
    int blocks = B / 4;                // 4 waves = 4 batches per block
    ep_detect_kernel<<<blocks, 128, 0, stream>>>(yr, yi, hr, hi, sd, out);
}